// minGRU_83159156785403
// MI455X (gfx1250) — compile-verified
//
#include <hip/hip_runtime.h>

typedef __attribute__((ext_vector_type(2))) float v2f;
typedef __attribute__((ext_vector_type(8))) float v8f;

#define LLEN 8192
#define TILE 256
#define NTILE (LLEN / TILE)

// ---- fast math: native v_exp_f32 / v_log_f32 / v_rcp_f32, no IEEE div chains
__device__ __forceinline__ float frcp_(float x) { return __builtin_amdgcn_rcpf(x); }
__device__ __forceinline__ float fexp_(float x) { return __expf(x); }
__device__ __forceinline__ float fsoftplus_(float x) {
  return (x > 15.0f) ? x : __logf(1.0f + __expf(x));
}
__device__ __forceinline__ float fsigmoid_(float x) {
  return frcp_(1.0f + __expf(-x));
}

// Issue 1 KB (256 floats) memory->LDS: 2x global_load_async_to_lds_b128,
// 16 B per lane per instruction. Tracked by ASYNCcnt.
__device__ __forceinline__ void async_tile_load(const float* base, unsigned voff_bytes,
                                                unsigned lds_bytes) {
  asm volatile(
      "global_load_async_to_lds_b128 %0, %1, %2\n\t"
      "global_load_async_to_lds_b128 %0, %1, %2 offset:512"
      :
      : "v"(lds_bytes), "v"(voff_bytes), "s"(base)
      : "memory");
}
__device__ __forceinline__ void wait_async_le4() {
  asm volatile("s_wait_asynccnt 0x4" ::: "memory");
}
__device__ __forceinline__ void wait_async_0() {
  asm volatile("s_wait_asynccnt 0x0" ::: "memory");
}

// One wave per output row; 256-position tiles = 16 chunks x 16 positions.
// Intra-chunk scans are triangular-matrix cumsums on the WMMA unit
// (4 chained v_wmma_f32_16x16x4_f32 per cumsum, two cumsums per tile).
// h/g tiles stream in via double-buffered async LDS loads.
__global__ __launch_bounds__(32) void mingru_wmma_scan(
    const float* __restrict__ x, float* __restrict__ out) {
  __shared__ __align__(16) float sH[2][TILE];   // raw h tile (memory order)
  __shared__ __align__(16) float sG[2][TILE];   // raw g tile (memory order)
  __shared__ __align__(16) float sLA[TILE];     // log(a), scan order
  __shared__ __align__(16) float sBB[TILE];     // b = z*h_tilde, scan order
  __shared__ __align__(16) float sU[TILE];      // b * exp(-LP)
  __shared__ __align__(16) float sO[TILE];      // outputs, scan order
  __shared__ __align__(16) float sCA[16];       // per-chunk A = P[15]
  __shared__ __align__(16) float sCB[16];       // per-chunk B = P[15]*C[15]

  const int lane = threadIdx.x;  // 0..31
  const int row  = blockIdx.x;   // 0..4095
  const int b    = row >> 9;
  const int c    = row & 511;
  const bool fwd = (c < 256);
  const int hch = fwd ? c : (c + 256);
  const int gch = fwd ? (c + 256) : (c + 512);
  const float* __restrict__ hP = x + (size_t)(b * 1024 + hch) * LLEN;
  const float* __restrict__ gP = x + (size_t)(b * 1024 + gch) * LLEN;
  float* __restrict__ oP = out + (size_t)row * LLEN;

  const int mn = lane & 15;  // A-row (M) == B-col (N) for this lane
  const int hl = lane >> 4;  // half-wave select
  const int k0 = hl * 2;     // K offset within a 4-wide slice (ISA A/B layout)

  // Loop-invariant triangular-ones A slices: T[m,k] = (k <= m)
  v2f Atri[4];
#pragma unroll
  for (int s = 0; s < 4; ++s) {
    Atri[s].x = ((4 * s + k0)     <= mn) ? 1.0f : 0.0f;
    Atri[s].y = ((4 * s + k0 + 1) <= mn) ? 1.0f : 0.0f;
  }

  const unsigned laneB = (unsigned)lane * 16u;  // 16 B per lane per async b128
  const unsigned ldsH[2] = {(unsigned)(size_t)&sH[0][0] + laneB,
                            (unsigned)(size_t)&sH[1][0] + laneB};
  const unsigned ldsG[2] = {(unsigned)(size_t)&sG[0][0] + laneB,
                            (unsigned)(size_t)&sG[1][0] + laneB};

  // Memory byte offset of this lane's slice of the tile block (memory order).
  // Backward rows read the block [LLEN-TILE-t0, LLEN-t0).
  auto memOff = [&](int t0) -> unsigned {
    const int e = fwd ? t0 : (LLEN - TILE - t0);
    return (unsigned)e * 4u + laneB;
  };

  // Preload tile 0 into buffer 0 (4 async b128 ops).
  async_tile_load(hP, memOff(0), ldsH[0]);
  async_tile_load(gP, memOff(0), ldsG[0]);

  float carry = 0.0f;  // wave-uniform scan state entering the current tile

  for (int tile = 0; tile < NTILE; ++tile) {
    const int t0  = tile * TILE;
    const int cur = tile & 1;
    const int t   = lane * 8;  // this lane's 8 contiguous scan positions

    // Kick off next tile's loads, then wait for current tile's group
    // (async loads complete in order: <=4 outstanding => current 4 are done).
    if (tile + 1 < NTILE) {
      async_tile_load(hP, memOff(t0 + TILE), ldsH[cur ^ 1]);
      async_tile_load(gP, memOff(t0 + TILE), ldsG[cur ^ 1]);
      wait_async_le4();
    } else {
      wait_async_0();
    }

    // ---------- Phase A: LDS reads (reverse indexing for backward rows) ----
    const float* sh = sH[cur];
    const float* sg = sG[cur];
    float hv[8], gv[8];
    if (fwd) {
      float4 a0 = *(const float4*)(sh + t);
      float4 a1 = *(const float4*)(sh + t + 4);
      float4 b0 = *(const float4*)(sg + t);
      float4 b1 = *(const float4*)(sg + t + 4);
      hv[0]=a0.x; hv[1]=a0.y; hv[2]=a0.z; hv[3]=a0.w;
      hv[4]=a1.x; hv[5]=a1.y; hv[6]=a1.z; hv[7]=a1.w;
      gv[0]=b0.x; gv[1]=b0.y; gv[2]=b0.z; gv[3]=b0.w;
      gv[4]=b1.x; gv[5]=b1.y; gv[6]=b1.z; gv[7]=b1.w;
    } else {
      // LDS index j holds scan position 255 - j
      float4 a0 = *(const float4*)(sh + (252 - t));
      float4 a1 = *(const float4*)(sh + (248 - t));
      float4 b0 = *(const float4*)(sg + (252 - t));
      float4 b1 = *(const float4*)(sg + (248 - t));
      hv[0]=a0.w; hv[1]=a0.z; hv[2]=a0.y; hv[3]=a0.x;
      hv[4]=a1.w; hv[5]=a1.z; hv[6]=a1.y; hv[7]=a1.x;
      gv[0]=b0.w; gv[1]=b0.z; gv[2]=b0.y; gv[3]=b0.x;
      gv[4]=b1.w; gv[5]=b1.z; gv[6]=b1.y; gv[7]=b1.x;
    }
    float la[8], bb[8];
#pragma unroll
    for (int i = 0; i < 8; ++i) {
      const float g = gv[i];
      const float h = hv[i];
      la[i] = -fsoftplus_(g);                               // log(1 - z)
      const float z  = fsigmoid_(g);
      const float ht = (h < 0.0f) ? fsigmoid_(h) : (h + 0.5f);
      bb[i] = z * ht;
    }
    *(float4*)(sLA + t)     = make_float4(la[0], la[1], la[2], la[3]);
    *(float4*)(sLA + t + 4) = make_float4(la[4], la[5], la[6], la[7]);
    *(float4*)(sBB + t)     = make_float4(bb[0], bb[1], bb[2], bb[3]);
    *(float4*)(sBB + t + 4) = make_float4(bb[4], bb[5], bb[6], bb[7]);

    // ---------- Phase B: LP = inclusive cumsum_k(la) via triangular WMMA ----
    v8f LP = {};
#pragma unroll
    for (int s = 0; s < 4; ++s) {
      v2f Bv = *(const v2f*)(sLA + mn * 16 + 4 * s + k0);
      LP = __builtin_amdgcn_wmma_f32_16x16x4_f32(false, Atri[s], false, Bv,
                                                 (short)0, LP, false, false);
    }

    // ---------- Phase C: P = exp(LP), u = b / P ----------
    float4 q0 = *(const float4*)(sBB + mn * 16 + 8 * hl);
    float4 q1 = *(const float4*)(sBB + mn * 16 + 8 * hl + 4);
    const float bbv[8] = {q0.x, q0.y, q0.z, q0.w, q1.x, q1.y, q1.z, q1.w};
    float P[8], U[8];
#pragma unroll
    for (int r = 0; r < 8; ++r) {
      P[r] = fexp_(LP[r]);
      U[r] = bbv[r] * frcp_(P[r]);   // = b * exp(-LP)
    }
    *(float4*)(sU + mn * 16 + 8 * hl)     = make_float4(U[0], U[1], U[2], U[3]);
    *(float4*)(sU + mn * 16 + 8 * hl + 4) = make_float4(U[4], U[5], U[6], U[7]);

    // ---------- Phase D: C = inclusive cumsum_k(u) via triangular WMMA ------
    v8f Cc = {};
#pragma unroll
    for (int s = 0; s < 4; ++s) {
      v2f Bv = *(const v2f*)(sU + mn * 16 + 4 * s + k0);
      Cc = __builtin_amdgcn_wmma_f32_16x16x4_f32(false, Atri[s], false, Bv,
                                                 (short)0, Cc, false, false);
    }

    // ---------- Phase E: chunk composition + cross-chunk serial scan --------
    if (lane >= 16) {            // these lanes hold m=15 in D-register r=7
      sCA[mn] = P[7];            // A_chunk = prod of the chunk's 16 a's
      sCB[mn] = P[7] * Cc[7];    // B_chunk = chunk-local scan end value
    }
    float4 ca0 = *(const float4*)(sCA);
    float4 ca1 = *(const float4*)(sCA + 4);
    float4 ca2 = *(const float4*)(sCA + 8);
    float4 ca3 = *(const float4*)(sCA + 12);
    float4 cb0 = *(const float4*)(sCB);
    float4 cb1 = *(const float4*)(sCB + 4);
    float4 cb2 = *(const float4*)(sCB + 8);
    float4 cb3 = *(const float4*)(sCB + 12);
    const float Ach[16] = {ca0.x, ca0.y, ca0.z, ca0.w, ca1.x, ca1.y, ca1.z, ca1.w,
                           ca2.x, ca2.y, ca2.z, ca2.w, ca3.x, ca3.y, ca3.z, ca3.w};
    const float Bch[16] = {cb0.x, cb0.y, cb0.z, cb0.w, cb1.x, cb1.y, cb1.z, cb1.w,
                           cb2.x, cb2.y, cb2.z, cb2.w, cb3.x, cb3.y, cb3.z, cb3.w};
    float cc  = carry;
    float myc = 0.0f;  // carry entering this lane's chunk (n = mn)
#pragma unroll
    for (int i = 0; i < 16; ++i) {
      if (i == mn) myc = cc;
      cc = fmaf(Ach[i], cc, Bch[i]);
    }
    carry = cc;  // wave-uniform carry into the next tile

    // out[m,n] = P[m]*(chunk_carry + C[m]); bounce via LDS for coalesced store
    float4 o0 = make_float4(P[0] * (myc + Cc[0]), P[1] * (myc + Cc[1]),
                            P[2] * (myc + Cc[2]), P[3] * (myc + Cc[3]));
    float4 o1 = make_float4(P[4] * (myc + Cc[4]), P[5] * (myc + Cc[5]),
                            P[6] * (myc + Cc[6]), P[7] * (myc + Cc[7]));
    *(float4*)(sO + mn * 16 + 8 * hl)     = o0;
    *(float4*)(sO + mn * 16 + 8 * hl + 4) = o1;

    float4 w0 = *(const float4*)(sO + t);
    float4 w1 = *(const float4*)(sO + t + 4);
    if (fwd) {
      *(float4*)(oP + t0 + t)     = w0;
      *(float4*)(oP + t0 + t + 4) = w1;
    } else {
      *(float4*)(oP + (LLEN - 4 - (t0 + t))) = make_float4(w0.w, w0.z, w0.y, w0.x);
      *(float4*)(oP + (LLEN - 8 - (t0 + t))) = make_float4(w1.w, w1.z, w1.y, w1.x);
    }
  }
}

extern "C" void kernel_launch(void* const* d_in, const int* in_sizes, int n_in,
                              void* d_out, int out_size, void* d_ws, size_t ws_size,
                              hipStream_t stream) {
  (void)in_sizes; (void)n_in; (void)out_size; (void)d_ws; (void)ws_size;
  const float* x = (const float*)d_in[0];
  float* out = (float*)d_out;
  // 8*512 = 4096 output rows; one wave (32 threads) per row.
  dim3 grid(4096), block(32);
  hipLaunchKernelGGL(mingru_wmma_scan, grid, block, 0, stream, x, out);
}